// GCN_trans_layer_65996467470506
// MI455X (gfx1250) — compile-verified
//
#include <hip/hip_runtime.h>
#include <hip/hip_bf16.h>
#include <stdint.h>

#define NFEAT 256
#define NHID  128

typedef __attribute__((ext_vector_type(2))) float v2f;
typedef __attribute__((ext_vector_type(8))) float v8f;

// --- deterministic per-element dropout hash (keep p = 0.9) ---
__device__ __forceinline__ uint32_t hash_u32(uint32_t x) {
    x ^= x >> 16; x *= 0x7feb352dU;
    x ^= x >> 15; x *= 0x846ca68bU;
    x ^= x >> 16;
    return x;
}
#define KEEP_THRESH 3865470566u   /* floor(0.9 * 2^32) */
#define INV_KEEP    1.1111111111111112f

// ============================================================
// Kernel 0: one-time transpose W[256][128] -> Wt[128][256]
// so B-fragments in the GEMM become single b64 loads.
// ============================================================
__global__ void __launch_bounds__(256)
gcn_transpose_w(const float* __restrict__ w, float* __restrict__ wt) {
    int idx = blockIdx.x * blockDim.x + threadIdx.x;   // [0, 256*128)
    if (idx < NFEAT * NHID) {
        int k = idx >> 7;        // 0..255
        int n = idx & (NHID-1);  // 0..127
        wt[n * NFEAT + k] = w[idx];
    }
}

// ============================================================
// Kernel 1: support = dropout(x @ W), fp32 WMMA 16x16x4
// block = 256 threads = 8 waves; block computes 16 rows x 128 cols.
// A from LDS (ds_load_2addr_b64), B from transposed W (global_load_b64).
// ============================================================
__global__ void __launch_bounds__(256)
gcn_gemm_dropout_wmma(const float* __restrict__ x,
                      const float* __restrict__ wt,   // [128][256] transposed
                      float* __restrict__ support) {
    __shared__ float xs[16 * NFEAT];   // 16 KB slab of x

    const int tid    = threadIdx.x;
    const int lane   = tid & 31;
    const int waveid = tid >> 5;            // 0..7 -> N tile
    const int m_base = blockIdx.x * 16;
    const int n_base = waveid * 16;

    // --- cooperative load: x[m_base : m_base+16, 0:256] -> LDS ---
    // 16*256 floats = 1024 float4; 256 threads x 4 float4 each
    {
        const float4* xg = reinterpret_cast<const float4*>(x + (size_t)m_base * NFEAT);
        float4* xl = reinterpret_cast<float4*>(xs);
        #pragma unroll
        for (int i = 0; i < 4; ++i) {
            int f = tid + i * 256;          // float4 index in [0,1024)
            xl[f] = xg[f];                  // row-major, contiguous slab
        }
    }
    __syncthreads();

    const int ksel  = (lane >> 4) << 1;     // 0 for lanes 0-15, 2 for lanes 16-31
    const int arow  = lane & 15;            // A-fragment row within tile
    const int gcol  = n_base + (lane & 15); // B/C column

    // per-lane contiguous K-stream through Wt: Wt[gcol][k0+ksel .. +1]
    const float* wcol = wt + (size_t)gcol * NFEAT + ksel;

    v8f acc = {};

    #pragma unroll 8
    for (int k0 = 0; k0 < NFEAT; k0 += 4) {
        // A fragment from LDS (8B-aligned: k0%4==0, ksel even)
        v2f a = *reinterpret_cast<const v2f*>(&xs[arow * NFEAT + k0 + ksel]);
        // B fragment: single 8B load from transposed W (L2/WGP$-resident)
        v2f b = *reinterpret_cast<const v2f*>(wcol + k0);
        acc = __builtin_amdgcn_wmma_f32_16x16x4_f32(
                  /*neg_a=*/false, a, /*neg_b=*/false, b,
                  /*c_mod=*/(short)0, acc,
                  /*reuse_a=*/false, /*reuse_b=*/false);
    }

    // --- epilogue: dropout + store support ---
    const int m_half = (lane >> 4) << 3;    // lanes 16-31 hold M=8..15
    #pragma unroll
    for (int r = 0; r < 8; ++r) {
        const int gm  = m_base + m_half + r;
        const uint32_t idx = (uint32_t)gm * NHID + (uint32_t)gcol;
        const float keep = (hash_u32(idx ^ 0x9E3779B9u) < KEEP_THRESH) ? INV_KEEP : 0.0f;
        support[(size_t)gm * NHID + gcol] = acc[r] * keep;
    }
}

// ============================================================
// Kernel 2: zero the output (harness poisons d_out)
// ============================================================
__global__ void __launch_bounds__(256)
gcn_zero(float4* __restrict__ out, int n4) {
    int i = blockIdx.x * blockDim.x + threadIdx.x;
    if (i < n4) out[i] = make_float4(0.f, 0.f, 0.f, 0.f);
}

// ============================================================
// Kernel 3: COO SpMM scatter-add. One wave (32 lanes) per edge,
// each lane handles 4 of the 128 features (float4 gather out of
// L2-resident support, 4x global_atomic_add_f32 scatter).
// ============================================================
__global__ void __launch_bounds__(256)
gcn_spmm_scatter(const float* __restrict__ support,
                 const float* __restrict__ vals,
                 const int*   __restrict__ rows,
                 const int*   __restrict__ cols,
                 float* __restrict__ out, int n_edges) {
    const int wave = (blockIdx.x * blockDim.x + threadIdx.x) >> 5;
    const int lane = threadIdx.x & 31;
    if (wave >= n_edges) return;

    const int   src = cols[wave];
    const int   dst = rows[wave];
    const float v   = vals[wave];

    const float4* sp = reinterpret_cast<const float4*>(support + (size_t)src * NHID);
    float4 m = sp[lane];                    // 32 lanes x 16B = 128 floats
    m.x *= v; m.y *= v; m.z *= v; m.w *= v;

    float* o = out + (size_t)dst * NHID + lane * 4;
    atomicAdd(o + 0, m.x);
    atomicAdd(o + 1, m.y);
    atomicAdd(o + 2, m.z);
    atomicAdd(o + 3, m.w);
}

// ============================================================
extern "C" void kernel_launch(void* const* d_in, const int* in_sizes, int n_in,
                              void* d_out, int out_size, void* d_ws, size_t ws_size,
                              hipStream_t stream) {
    const float* x     = (const float*)d_in[0];   // [N, 256]
    const float* w     = (const float*)d_in[1];   // [256, 128]
    const float* vals  = (const float*)d_in[2];   // [E]
    const int*   rows  = (const int*)  d_in[3];   // [E]
    const int*   cols  = (const int*)  d_in[4];   // [E]

    const int n_nodes = in_sizes[0] / NFEAT;      // 100000
    const int n_edges = in_sizes[2];              // 1600000

    // workspace layout: support [N,128] at 0; Wt [128][256] after it
    float* support = (float*)d_ws;
    float* wt      = support + (size_t)n_nodes * NHID;
    float* out     = (float*)d_out;

    // 0) transpose W (one-time, 32K elements)
    gcn_transpose_w<<<(NFEAT * NHID + 255) / 256, 256, 0, stream>>>(w, wt);

    // 1) GEMM + dropout (16 rows per block; 100000 divides by 16 -> 6250)
    gcn_gemm_dropout_wmma<<<n_nodes / 16, 256, 0, stream>>>(x, wt, support);

    // 2) zero output
    const int n4 = out_size / 4;
    gcn_zero<<<(n4 + 255) / 256, 256, 0, stream>>>((float4*)out, n4);

    // 3) scatter-add SpMM: 8 edges (waves) per 256-thread block
    const int blocks = (n_edges + 7) / 8;
    gcn_spmm_scatter<<<blocks, 256, 0, stream>>>(support, vals, rows, cols, out, n_edges);
}